// SamplesLoss_20813411516762
// MI455X (gfx1250) — compile-verified
//
#include <hip/hip_runtime.h>
#include <math.h>

// ---------------- Problem constants (from reference) ----------------
#define B_SZ  8
#define NPTS  2048
#define BN    (B_SZ * NPTS)

#define LOG2E 1.4426950408889634f
#define LN2   0.6931471805599453f

typedef __attribute__((ext_vector_type(2))) float v2f;
typedef __attribute__((ext_vector_type(8))) float v8f;

// LDS layout (float indices). Two packed float2 "planes" so each lane's
// per-tile B-fragment fetch is a single branchless ds_load_b64:
//   plane0[j] = (qx[j], qy[j])   -> read by lanes 0-15  (K0,K1 rows)
//   plane1[j] = (qz[j], 0.0f)    -> read by lanes 16-31 (K2,K3 rows)
// plane1 is offset by +32 floats so the two half-waves hit disjoint
// LDS bank groups (64 banks x 4B).
#define PK0_OFF 0
#define PK1_OFF (2 * NPTS + 32)
#define AL_OFF  (2 * (2 * NPTS + 32))
#define LDS_FLOATS (AL_OFF + NPTS + 16)

// ---------------------------------------------------------------------------
// softmin over columns with on-the-fly cost via fp32 WMMA, two-pass LSE.
//   out_i = -eps * log sum_j exp(h_j - C_ij/eps),  C_ij = psq_i + qsq_j - p.q
// Rearranged (base-2):
//   alpha2_j = (h_j - qsq_j/eps)*log2e      (staged in LDS)
//   d2_ij    = (p_i.q_j)/eps*log2e          (WMMA, scale folded into A)
//   out_i    = psq_i - eps*ln2*( M_i + log2 sum_j 2^(alpha2_j + d2_ij - M_i) )
// Pass 1 computes M_i (max only, no TRANS ops); pass 2 does ONE exp2 per
// element. The WMMA dot is recomputed in pass 2 — points live in LDS, so a
// second v_wmma is far cheaper than the rescale exp it replaces.
// One wave = 16 rows (one WMMA A fragment). Block = 4 waves = 64 rows.
// ---------------------------------------------------------------------------
__launch_bounds__(128)
__global__ void sinkhorn_softmin_wmma(
    const float* __restrict__ P,    // (B, N, 3) row points
    const float* __restrict__ Q,    // (B, M, 3) column points
    const float* __restrict__ Psq,  // (B, N)  0.5*|p|^2
    const float* __restrict__ Qsq,  // (B, M)  0.5*|q|^2
    const float* __restrict__ H,    // (B, M)  dual "h" vector
    float* __restrict__ Out,        // (B, N)  result (optionally averaged)
    float eps, int avg)
{
    __shared__ float lds[LDS_FLOATS];

    const int b       = blockIdx.y;
    const int tid     = threadIdx.x;
    const float inv_eps = 1.0f / eps;

    // ---- stage packed column planes + alpha into LDS ----
    const float* Qb   = Q   + (size_t)b * NPTS * 3;
    const float* Qsqb = Qsq + b * NPTS;
    const float* Hb   = H   + b * NPTS;
    for (int j = tid; j < NPTS; j += 128) {
        float qx = Qb[3 * j + 0];
        float qy = Qb[3 * j + 1];
        float qz = Qb[3 * j + 2];
        lds[PK0_OFF + 2 * j + 0] = qx;
        lds[PK0_OFF + 2 * j + 1] = qy;
        lds[PK1_OFF + 2 * j + 0] = qz;
        lds[PK1_OFF + 2 * j + 1] = 0.0f;            // K=3 padding row
        lds[AL_OFF + j] = (Hb[j] - Qsqb[j] * inv_eps) * LOG2E;
    }
    __syncthreads();

    const int wave = tid >> 5;
    const int lane = tid & 31;
    const int n    = lane & 15;   // column-within-tile / row-within-fragment
    const int hi   = lane >> 4;   // upper half-wave flag
    const int r0   = blockIdx.x * 64 + wave * 16;

    // A fragment, 16x4 fp32 (ISA 7.12.2): lane l     -> A[l][0],A[l][1]
    //                                     lane 16+l  -> A[l][2],A[l][3]
    // K = (x, y, z, 0); fold (1/eps)*log2e into A.
    // Unconditional loads + register selects: no exec-mask divergence.
    const float* Pr = P + (size_t)b * NPTS * 3 + (size_t)(r0 + n) * 3;
    const float px = Pr[0], py = Pr[1], pz = Pr[2];
    const float ascale = inv_eps * LOG2E;
    v2f afrag;
    afrag.x = (hi ? pz : px) * ascale;
    afrag.y = (hi ? 0.0f : py * ascale);

    // branchless per-lane B-fragment base pointer (computed once)
    const v2f*   bplane = (const v2f*)&lds[(hi ? PK1_OFF : PK0_OFF) + 2 * n];
    const float* alp    = &lds[AL_OFF + n];

    // ================= pass 1: row maxima (VALU only) =================
    float m[8];
#pragma unroll
    for (int v = 0; v < 8; v++) m[v] = -3.0e38f;

#pragma unroll 2
    for (int c = 0; c < NPTS; c += 16) {
        v2f  bfrag = bplane[c];            // single ds_load_b64
        v8f  acc   = {0.f, 0.f, 0.f, 0.f, 0.f, 0.f, 0.f, 0.f};
        acc = __builtin_amdgcn_wmma_f32_16x16x4_f32(
                  false, afrag, false, bfrag, (short)0, acc, false, false);
        const float al = alp[c];           // per-column alpha, same in both half-waves
#pragma unroll
        for (int v = 0; v < 8; v++)
            m[v] = fmaxf(m[v], acc[v] + al);
    }

    // merge max across the 16 lanes of each half-wave (max-only tree)
#pragma unroll
    for (int v = 0; v < 8; v++) {
#pragma unroll
        for (int off = 1; off < 16; off <<= 1)
            m[v] = fmaxf(m[v], __shfl_xor(m[v], off, 32));
    }
    // now every lane of a half-wave holds the true row max M for its 8 rows

    // ================= pass 2: one exp2 per element =================
    float s[8];
#pragma unroll
    for (int v = 0; v < 8; v++) s[v] = 0.0f;

#pragma unroll 2
    for (int c = 0; c < NPTS; c += 16) {
        v2f  bfrag = bplane[c];
        v8f  acc   = {0.f, 0.f, 0.f, 0.f, 0.f, 0.f, 0.f, 0.f};
        acc = __builtin_amdgcn_wmma_f32_16x16x4_f32(
                  false, afrag, false, bfrag, (short)0, acc, false, false);
        const float al = alp[c];
#pragma unroll
        for (int v = 0; v < 8; v++)
            s[v] += __builtin_amdgcn_exp2f(acc[v] + al - m[v]);  // <= 1, safe
    }

    // ---- add-only merge across lanes, then write 16 rows ----
#pragma unroll
    for (int v = 0; v < 8; v++) {
#pragma unroll
        for (int off = 1; off < 16; off <<= 1)
            s[v] += __shfl_xor(s[v], off, 32);
        if (n == v) {                     // one lane per half-wave writes
            int   row = r0 + v + 8 * hi;
            float L2  = m[v] + __builtin_amdgcn_logf(s[v]);   // log2 domain
            float res = Psq[b * NPTS + row] - eps * LN2 * L2;
            float* o  = Out + b * NPTS + row;
            *o = avg ? 0.5f * (*o + res) : res;
        }
    }
}

// h[i] = base + g[i]/eps   (g may be null -> constant)
__global__ void prep_h_kernel(const float* __restrict__ g, float* __restrict__ h,
                              float inv_eps, float base, int count)
{
    int i = blockIdx.x * blockDim.x + threadIdx.x;
    if (i < count) {
        float v = base;
        if (g) v += g[i] * inv_eps;
        h[i] = v;
    }
}

__global__ void sqnorm_kernel(const float* __restrict__ pts, float* __restrict__ sq,
                              int count)
{
    int i = blockIdx.x * blockDim.x + threadIdx.x;
    if (i < count) {
        float x = pts[3 * i], y = pts[3 * i + 1], z = pts[3 * i + 2];
        sq[i] = 0.5f * (x * x + y * y + z * z);
    }
}

// loss[b] = (1/N) sum_i (f_ba - f_aa) + (1/M) sum_j (g_ab - g_bb)
__global__ void reduce_loss_kernel(const float* __restrict__ f_ba,
                                   const float* __restrict__ f_aa,
                                   const float* __restrict__ g_ab,
                                   const float* __restrict__ g_bb,
                                   float* __restrict__ out)
{
    const int b   = blockIdx.x;
    const int tid = threadIdx.x;  // 256 threads
    float t = 0.0f;
    for (int i = tid; i < NPTS; i += 256) {
        int idx = b * NPTS + i;
        t += (f_ba[idx] - f_aa[idx]) + (g_ab[idx] - g_bb[idx]);
    }
    for (int off = 16; off > 0; off >>= 1) t += __shfl_down(t, off, 32);
    __shared__ float acc[8];
    if ((tid & 31) == 0) acc[tid >> 5] = t;
    __syncthreads();
    if (tid == 0) {
        float sum = 0.0f;
        for (int w = 0; w < 8; w++) sum += acc[w];
        out[b] = sum * (1.0f / (float)NPTS);
    }
}

// ---------------------------------------------------------------------------
extern "C" void kernel_launch(void* const* d_in, const int* in_sizes, int n_in,
                              void* d_out, int out_size, void* d_ws, size_t ws_size,
                              hipStream_t stream)
{
    const float* x = (const float*)d_in[0];   // (8, 2048, 3) f32
    const float* y = (const float*)d_in[1];   // (8, 2048, 3) f32
    float* out = (float*)d_out;               // (8,) f32
    float* ws  = (float*)d_ws;

    // workspace layout: 10 * BN floats = 640 KB
    float* xsq   = ws + 0 * BN;
    float* ysq   = ws + 1 * BN;
    float* f_aa  = ws + 2 * BN;
    float* g_bb  = ws + 3 * BN;
    float* f_ba  = ws + 4 * BN;
    float* g_ab  = ws + 5 * BN;
    float* hy_ab = ws + 6 * BN;   // alpha-source over y columns (-> f_ba)
    float* hx_ba = ws + 7 * BN;   // over x columns (-> g_ab)
    float* hx_aa = ws + 8 * BN;   // over x columns (-> f_aa)
    float* hy_bb = ws + 9 * BN;   // over y columns (-> g_bb)

    const float a_log = -logf((float)NPTS);   // == b_log (uniform weights)

    // geomloss epsilon schedule: [diam^p] + exp(arange(p ln diam, p ln blur, p ln scal)) + [blur^p]
    const double p = 2.0, diameter = 2.0, blur = 0.05, scaling = 0.5;
    float eps_list[16];
    int n_eps = 0;
    eps_list[n_eps++] = (float)pow(diameter, p);
    for (double e = p * log(diameter); e > p * log(blur); e += p * log(scaling))
        eps_list[n_eps++] = (float)exp(e);
    eps_list[n_eps++] = (float)pow(blur, p);

    dim3 sm_grid(NPTS / 64, B_SZ), sm_blk(128);
    dim3 ew_grid(BN / 256),        ew_blk(256);

    sqnorm_kernel<<<ew_grid, ew_blk, 0, stream>>>(x, xsq, BN);
    sqnorm_kernel<<<ew_grid, ew_blk, 0, stream>>>(y, ysq, BN);

    // ---- initialization at eps0 (h = a_log / b_log constants) ----
    const float eps0 = eps_list[0];
    prep_h_kernel<<<ew_grid, ew_blk, 0, stream>>>(nullptr, hx_aa, 0.f, a_log, BN);
    prep_h_kernel<<<ew_grid, ew_blk, 0, stream>>>(nullptr, hy_bb, 0.f, a_log, BN);
    sinkhorn_softmin_wmma<<<sm_grid, sm_blk, 0, stream>>>(x, x, xsq, xsq, hx_aa, f_aa, eps0, 0); // C_xx
    sinkhorn_softmin_wmma<<<sm_grid, sm_blk, 0, stream>>>(y, y, ysq, ysq, hy_bb, g_bb, eps0, 0); // C_yy
    sinkhorn_softmin_wmma<<<sm_grid, sm_blk, 0, stream>>>(y, x, ysq, xsq, hx_aa, g_ab, eps0, 0); // C_yx
    sinkhorn_softmin_wmma<<<sm_grid, sm_blk, 0, stream>>>(x, y, xsq, ysq, hy_bb, f_ba, eps0, 0); // C_xy

    // ---- symmetric Sinkhorn loop with eps scaling (averaged updates) ----
    for (int it = 0; it < n_eps; it++) {
        const float eps = eps_list[it], inv_eps = 1.0f / eps;
        // snapshot old duals into h buffers first (removes RAW hazards)
        prep_h_kernel<<<ew_grid, ew_blk, 0, stream>>>(g_ab, hy_ab, inv_eps, a_log, BN);
        prep_h_kernel<<<ew_grid, ew_blk, 0, stream>>>(f_ba, hx_ba, inv_eps, a_log, BN);
        prep_h_kernel<<<ew_grid, ew_blk, 0, stream>>>(f_aa, hx_aa, inv_eps, a_log, BN);
        prep_h_kernel<<<ew_grid, ew_blk, 0, stream>>>(g_bb, hy_bb, inv_eps, a_log, BN);
        sinkhorn_softmin_wmma<<<sm_grid, sm_blk, 0, stream>>>(x, y, xsq, ysq, hy_ab, f_ba, eps, 1);
        sinkhorn_softmin_wmma<<<sm_grid, sm_blk, 0, stream>>>(y, x, ysq, xsq, hx_ba, g_ab, eps, 1);
        sinkhorn_softmin_wmma<<<sm_grid, sm_blk, 0, stream>>>(x, x, xsq, xsq, hx_aa, f_aa, eps, 1);
        sinkhorn_softmin_wmma<<<sm_grid, sm_blk, 0, stream>>>(y, y, ysq, ysq, hy_bb, g_bb, eps, 1);
    }

    // ---- final detached extrapolation at last eps (overwrite, no averaging) ----
    const float epsl = eps_list[n_eps - 1], invl = 1.0f / epsl;
    prep_h_kernel<<<ew_grid, ew_blk, 0, stream>>>(g_ab, hy_ab, invl, a_log, BN);
    prep_h_kernel<<<ew_grid, ew_blk, 0, stream>>>(f_ba, hx_ba, invl, a_log, BN);
    prep_h_kernel<<<ew_grid, ew_blk, 0, stream>>>(f_aa, hx_aa, invl, a_log, BN);
    prep_h_kernel<<<ew_grid, ew_blk, 0, stream>>>(g_bb, hy_bb, invl, a_log, BN);
    sinkhorn_softmin_wmma<<<sm_grid, sm_blk, 0, stream>>>(x, y, xsq, ysq, hy_ab, f_ba, epsl, 0);
    sinkhorn_softmin_wmma<<<sm_grid, sm_blk, 0, stream>>>(y, x, ysq, xsq, hx_ba, g_ab, epsl, 0);
    sinkhorn_softmin_wmma<<<sm_grid, sm_blk, 0, stream>>>(x, x, xsq, xsq, hx_aa, f_aa, epsl, 0);
    sinkhorn_softmin_wmma<<<sm_grid, sm_blk, 0, stream>>>(y, y, ysq, ysq, hy_bb, g_bb, epsl, 0);

    reduce_loss_kernel<<<dim3(B_SZ), dim3(256), 0, stream>>>(f_ba, f_aa, g_ab, g_bb, out);
}